// GCN_55224689492024
// MI455X (gfx1250) — compile-verified
//
#include <hip/hip_runtime.h>
#include <stdint.h>

#define N_NODES   100000
#define N_EDGES   3200000
#define N_FEAT    512
#define N_HID     128
#define N_CLASS   40
#define N_CLS_PAD 48

typedef __attribute__((ext_vector_type(2))) float v2f;
typedef __attribute__((ext_vector_type(8))) float v8f;

// ---------------------------------------------------------------------------
// CDNA5 async global->LDS copy (ASYNCcnt-tracked), 16B per lane.
// Builtin prototype (from hipcc diagnostic): param0 = v4i addrspace(1)*,
// i.e. (global_src, lds_dst, imm offset, imm cpol).
// ---------------------------------------------------------------------------
typedef __attribute__((__vector_size__(16))) int v4i_t;
typedef __attribute__((address_space(1))) v4i_t* gv4i_p;
typedef __attribute__((address_space(3))) v4i_t* lv4i_p;

__device__ __forceinline__ void async_copy_f4(float* lds_dst, const float* gsrc) {
#if __has_builtin(__builtin_amdgcn_global_load_async_to_lds_b128)
  __builtin_amdgcn_global_load_async_to_lds_b128(
      (gv4i_p)(v4i_t*)(uintptr_t)gsrc,
      (lv4i_p)(v4i_t*)(uintptr_t)lds_dst,
      /*offset=*/0, /*cpol=*/0);
#else
  unsigned           laddr = (unsigned)(uintptr_t)lds_dst;
  unsigned long long gaddr = (unsigned long long)(uintptr_t)gsrc;
  asm volatile("global_load_async_to_lds_b128 %0, %1, off"
               :: "v"(laddr), "v"(gaddr) : "memory");
#endif
}

__device__ __forceinline__ void wait_async0() {
#if __has_builtin(__builtin_amdgcn_s_wait_asynccnt)
  __builtin_amdgcn_s_wait_asynccnt(0);
#else
  asm volatile("s_wait_asynccnt 0x0" ::: "memory");
#endif
}

// ---------------------------------------------------------------------------
// Kernel 1: T1 = X @ W1   (fp32 WMMA 16x16x4)
// Block = 64(M) x 128(N) output tile, 8 waves, wave w owns N-tile w and loops
// 4 M-subtiles (B fragment reused x4). K chunked by 32 with double-buffered
// async global->LDS staging of A (64x32) and B (32x128).
// ---------------------------------------------------------------------------
__global__ __launch_bounds__(256) void gcn_gemm1_wmma(
    const float* __restrict__ X, const float* __restrict__ W1,
    float* __restrict__ T1) {
  __shared__ float As[2][64 * 32];    // 2 x 8 KB
  __shared__ float Bs[2][32 * 128];   // 2 x 16 KB
  const int m0  = blockIdx.x * 64;
  const int tid = threadIdx.x;

  // staging coordinates (float4 granularity)
  const int ar0 = (tid * 2) >> 3;           // As: 512 float4, 2/thread
  const int ac0 = ((tid * 2) & 7) * 4;
  int ag0 = m0 + ar0;      if (ag0 >= N_NODES) ag0 = 0;   // clamp OOB rows

  auto stage = [&](int buf, int k0) {
    // A tile: row ar0, two adjacent 16B column chunks
    {
      const float* s0 = X + (size_t)ag0 * N_FEAT + k0 + ac0;
      async_copy_f4(&As[buf][ar0 * 32 + ac0],     s0);
      async_copy_f4(&As[buf][ar0 * 32 + ac0 + 4], s0 + 4);
    }
    // B tile: 4 float4 per thread
#pragma unroll
    for (int i = 0; i < 4; ++i) {
      int idx = tid + i * 256;                // 0..1023
      int k   = idx >> 5;
      int c4  = (idx & 31) * 4;
      async_copy_f4(&Bs[buf][k * 128 + c4],
                    W1 + (size_t)(k0 + k) * N_HID + c4);
    }
  };

  const int wave = tid >> 5;          // 0..7  -> N tile
  const int lane = tid & 31;
  const int n0   = wave * 16;
  const int mrow = lane & 15;
  const int kb   = (lane >> 4) * 2;   // lanes 0-15 -> K{0,1}; 16-31 -> K{2,3}
  const int ncol = lane & 15;

  v8f acc[4] = {};

  stage(0, 0);
  wait_async0();
  __syncthreads();

  int buf = 0;
  for (int k0 = 0; k0 < N_FEAT; k0 += 32) {
    if (k0 + 32 < N_FEAT) stage(buf ^ 1, k0 + 32);   // overlap copy w/ compute

#pragma unroll
    for (int kk = 0; kk < 32; kk += 4) {
      v2f b;
      b.x = Bs[buf][(kk + kb)     * 128 + n0 + ncol];
      b.y = Bs[buf][(kk + kb + 1) * 128 + n0 + ncol];
#pragma unroll
      for (int mt = 0; mt < 4; ++mt) {
        v2f a;
        a.x = As[buf][(mt * 16 + mrow) * 32 + kk + kb];
        a.y = As[buf][(mt * 16 + mrow) * 32 + kk + kb + 1];
        acc[mt] = __builtin_amdgcn_wmma_f32_16x16x4_f32(
            false, a, false, b, (short)0, acc[mt], false, false);
      }
    }

    wait_async0();        // next chunk's async copies done
    __syncthreads();
    buf ^= 1;
  }

#pragma unroll
  for (int mt = 0; mt < 4; ++mt) {
#pragma unroll
    for (int r = 0; r < 8; ++r) {
      int row = m0 + mt * 16 + r + (lane >> 4) * 8;
      if (row < N_NODES)
        T1[(size_t)row * N_HID + n0 + ncol] = acc[mt][r];
    }
  }
}

// ---------------------------------------------------------------------------
// Zero-fill
// ---------------------------------------------------------------------------
__global__ void gcn_zero(float* __restrict__ p, long n) {
  long i = (long)blockIdx.x * blockDim.x + threadIdx.x;
  if (i < n) p[i] = 0.0f;
}

// ---------------------------------------------------------------------------
// Kernel 2: scatter-add SpMM, D=128. One wave per edge, lane = float4 chunk.
// T1 (51.2 MB) and H (51.2 MB) stay resident in the 192 MB L2.
// ---------------------------------------------------------------------------
__global__ __launch_bounds__(256) void gcn_spmm128(
    const int* __restrict__ src, const int* __restrict__ dst,
    const float* __restrict__ ew, const float* __restrict__ Xin,
    float* __restrict__ Y) {
  long t = (long)blockIdx.x * blockDim.x + threadIdx.x;
  long e = t >> 5;
  if (e >= N_EDGES) return;
  int c0 = (int)(t & 31) * 4;
  int s = src[e], d = dst[e];
  float w = ew[e];
  const float4 v = *(const float4*)(Xin + (size_t)s * N_HID + c0);
  float* yp = Y + (size_t)d * N_HID + c0;
  atomicAdd(yp + 0, w * v.x);
  atomicAdd(yp + 1, w * v.y);
  atomicAdd(yp + 2, w * v.z);
  atomicAdd(yp + 3, w * v.w);
}

// ---------------------------------------------------------------------------
// Kernel 3: H = dropout(relu(H + b1)) — exact JAX Threefry2x32, key=(0,42).
// Flat counter [0, 12.8M) split in halves; block j encrypts (j, j+half).
// ---------------------------------------------------------------------------
__device__ __forceinline__ uint32_t rotl32(uint32_t x, int r) {
  return (x << r) | (x >> (32 - r));
}

__global__ __launch_bounds__(256) void gcn_post1(
    float* __restrict__ H, const float* __restrict__ b1) {
  const long half = (long)N_NODES * N_HID / 2;   // 6,400,000
  long i = (long)blockIdx.x * blockDim.x + threadIdx.x;
  if (i >= half) return;

  const uint32_t k0 = 0u, k1 = 42u;
  const uint32_t ks2 = k0 ^ k1 ^ 0x1BD11BDAu;
  uint32_t x0 = (uint32_t)i + k0;
  uint32_t x1 = (uint32_t)(i + half) + k1;
#define TFR(r) { x0 += x1; x1 = rotl32(x1, (r)); x1 ^= x0; }
  TFR(13) TFR(15) TFR(26) TFR(6)   x0 += k1;  x1 += ks2 + 1u;
  TFR(17) TFR(29) TFR(16) TFR(24)  x0 += ks2; x1 += k0 + 2u;
  TFR(13) TFR(15) TFR(26) TFR(6)   x0 += k0;  x1 += k1 + 3u;
  TFR(17) TFR(29) TFR(16) TFR(24)  x0 += k1;  x1 += ks2 + 4u;
  TFR(13) TFR(15) TFR(26) TFR(6)   x0 += ks2; x1 += k0 + 5u;
#undef TFR

  uint32_t u0b = (x0 >> 9) | 0x3f800000u;
  uint32_t u1b = (x1 >> 9) | 0x3f800000u;
  float u0, u1;
  __builtin_memcpy(&u0, &u0b, 4);
  __builtin_memcpy(&u1, &u1b, 4);
  u0 -= 1.0f; u1 -= 1.0f;

  int c = (int)(i & (N_HID - 1));   // half % 128 == 0 -> same column
  float h0 = fmaxf(H[i] + b1[c], 0.0f);
  float h1 = fmaxf(H[i + half] + b1[c], 0.0f);
  H[i]        = (u0 < 0.5f) ? h0 * 2.0f : 0.0f;   // keep_p = 0.5
  H[i + half] = (u1 < 0.5f) ? h1 * 2.0f : 0.0f;
}

// ---------------------------------------------------------------------------
// Kernel 4: L = H @ W2   (fp32 WMMA, N padded 40 -> 48 in LDS, 3 waves/block)
// ---------------------------------------------------------------------------
__global__ __launch_bounds__(96) void gcn_gemm2_wmma(
    const float* __restrict__ H, const float* __restrict__ W2,
    float* __restrict__ L) {
  __shared__ float Hs[16 * N_HID];           // 8 KB
  __shared__ float W2s[N_HID * N_CLS_PAD];   // 24 KB
  const int m0  = blockIdx.x * 16;
  const int tid = threadIdx.x;

  for (int idx = tid; idx < 16 * N_HID; idx += 96)
    Hs[idx] = H[(size_t)m0 * N_HID + idx];
  for (int idx = tid; idx < N_HID * N_CLS_PAD; idx += 96) {
    int k = idx / N_CLS_PAD, c = idx % N_CLS_PAD;
    W2s[idx] = (c < N_CLASS) ? W2[k * N_CLASS + c] : 0.0f;
  }
  __syncthreads();

  const int wave = tid >> 5;          // 0..2
  const int lane = tid & 31;
  const int n0   = wave * 16;
  const int mrow = lane & 15;
  const int kb   = (lane >> 4) * 2;
  const int ncol = lane & 15;

  v8f acc = {};
#pragma unroll
  for (int k0 = 0; k0 < N_HID; k0 += 4) {
    v2f a, b;
    a.x = Hs[mrow * N_HID + k0 + kb];
    a.y = Hs[mrow * N_HID + k0 + kb + 1];
    b.x = W2s[(k0 + kb)     * N_CLS_PAD + n0 + ncol];
    b.y = W2s[(k0 + kb + 1) * N_CLS_PAD + n0 + ncol];
    acc = __builtin_amdgcn_wmma_f32_16x16x4_f32(
        false, a, false, b, (short)0, acc, false, false);
  }

  int col = n0 + ncol;
  if (col < N_CLASS) {
#pragma unroll
    for (int r = 0; r < 8; ++r) {
      int row = r + (lane >> 4) * 8;
      L[(size_t)(m0 + row) * N_CLASS + col] = acc[r];
    }
  }
}

// ---------------------------------------------------------------------------
// Kernel 5: scatter-add SpMM, D=40. 10 float4 chunks per edge.
// ---------------------------------------------------------------------------
__global__ __launch_bounds__(256) void gcn_spmm40(
    const int* __restrict__ src, const int* __restrict__ dst,
    const float* __restrict__ ew, const float* __restrict__ L,
    float* __restrict__ Y) {
  long t = (long)blockIdx.x * blockDim.x + threadIdx.x;
  long e = t / 10;
  if (e >= N_EDGES) return;
  int c0 = (int)(t - e * 10) * 4;
  int s = src[e], d = dst[e];
  float w = ew[e];
  const float4 v = *(const float4*)(L + (size_t)s * N_CLASS + c0);
  float* yp = Y + (size_t)d * N_CLASS + c0;
  atomicAdd(yp + 0, w * v.x);
  atomicAdd(yp + 1, w * v.y);
  atomicAdd(yp + 2, w * v.z);
  atomicAdd(yp + 3, w * v.w);
}

// ---------------------------------------------------------------------------
// Kernel 6: out = log_softmax(Yagg + b2). One wave32 per node; lane covers
// classes {lane, lane+32}; shfl_xor tree reductions.
// ---------------------------------------------------------------------------
__global__ __launch_bounds__(256) void gcn_logsoftmax(
    const float* __restrict__ Yagg, const float* __restrict__ b2,
    float* __restrict__ out) {
  int wid = (int)(((long)blockIdx.x * blockDim.x + threadIdx.x) >> 5);
  if (wid >= N_NODES) return;
  int lane = threadIdx.x & 31;
  const float* row = Yagg + (size_t)wid * N_CLASS;

  float v0 = row[lane] + b2[lane];
  float v1 = (lane < 8) ? (row[32 + lane] + b2[32 + lane]) : -3.402823466e38f;

  float m = fmaxf(v0, v1);
#pragma unroll
  for (int off = 16; off >= 1; off >>= 1) m = fmaxf(m, __shfl_xor(m, off, 32));
  float s = expf(v0 - m) + ((lane < 8) ? expf(v1 - m) : 0.0f);
#pragma unroll
  for (int off = 16; off >= 1; off >>= 1) s += __shfl_xor(s, off, 32);
  float lse = m + logf(s);

  float* o = out + (size_t)wid * N_CLASS;
  o[lane] = v0 - lse;
  if (lane < 8) o[32 + lane] = v1 - lse;
}

// ---------------------------------------------------------------------------
extern "C" void kernel_launch(void* const* d_in, const int* in_sizes, int n_in,
                              void* d_out, int out_size, void* d_ws, size_t ws_size,
                              hipStream_t stream) {
  const float* feature = (const float*)d_in[0];
  const int*   esrc    = (const int*)d_in[1];
  const int*   edst    = (const int*)d_in[2];
  const float* ew      = (const float*)d_in[3];
  const float* W1      = (const float*)d_in[4];
  const float* b1      = (const float*)d_in[5];
  const float* W2      = (const float*)d_in[6];
  const float* b2      = (const float*)d_in[7];
  float* out = (float*)d_out;

  float* ws = (float*)d_ws;
  float* T1   = ws;                      // 12.8M floats : X@W1
  float* H    = ws + 12800000;           // 12.8M floats : aggregated hidden
  float* L    = ws + 25600000;           //  4.0M floats : H@W2
  float* Yagg = ws + 29600000;           //  4.0M floats : aggregated logits

  // 1) T1 = X @ W1  (fp32 WMMA, async-LDS double-buffered)
  gcn_gemm1_wmma<<<(N_NODES + 63) / 64, 256, 0, stream>>>(feature, W1, T1);

  // 2) H = segment_sum(T1[src] * w -> dst)
  {
    long n = (long)N_NODES * N_HID;
    gcn_zero<<<(int)((n + 255) / 256), 256, 0, stream>>>(H, n);
  }
  gcn_spmm128<<<(int)(((long)N_EDGES * 32) / 256), 256, 0, stream>>>(
      esrc, edst, ew, T1, H);

  // 3) H = dropout(relu(H + b1))
  gcn_post1<<<(int)(((long)N_NODES * N_HID / 2 + 255) / 256), 256, 0, stream>>>(H, b1);

  // 4) L = H @ W2  (fp32 WMMA, padded N)
  gcn_gemm2_wmma<<<N_NODES / 16, 96, 0, stream>>>(H, W2, L);

  // 5) Yagg = segment_sum(L[src] * w -> dst)
  {
    long n = (long)N_NODES * N_CLASS;
    gcn_zero<<<(int)((n + 255) / 256), 256, 0, stream>>>(Yagg, n);
  }
  gcn_spmm40<<<(int)(((long)N_EDGES * 10 + 255) / 256), 256, 0, stream>>>(
      esrc, edst, ew, L, Yagg);

  // 6) out = log_softmax(Yagg + b2)
  gcn_logsoftmax<<<(int)(((long)N_NODES * 32 + 255) / 256), 256, 0, stream>>>(
      Yagg, b2, out);
}